// SSIMLoss_56736517980943
// MI455X (gfx1250) — compile-verified
//
#include <hip/hip_runtime.h>

typedef float v2f __attribute__((ext_vector_type(2)));
typedef float v8f __attribute__((ext_vector_type(8)));

#define Hh 512
#define Ww 512
#define OUTW 506
#define STRIPW 128      // output columns per block
#define VCOLS 134       // STRIPW + 6 halo columns of vertical sums
#define VPAD 138        // padded leading dim (bank-friendly, covers k=22/23 reads)
#define ROWS_PER_BLOCK 64
#define NBATCH 4        // 4 batches of 16 output rows

__global__ __launch_bounds__(128) void ssim_main(const float* __restrict__ pred,
                                                 const float* __restrict__ targ,
                                                 float* __restrict__ accum) {
    __shared__ __align__(16) float vsum[5][16][VPAD];   // vertical 7-sums of 5 stats
    __shared__ float ring[8][VCOLS + 2][2];             // 8-row history of (p,t)
    __shared__ float red[128];

    const int tid   = threadIdx.x;
    const int xb    = blockIdx.x * STRIPW;              // 0,128,256,384
    const int yb    = blockIdx.y * ROWS_PER_BLOCK;      // 0..448
    const int plane = blockIdx.z;                       // 0..95
    const float* P = pred + (size_t)plane * (Hh * Ww);
    const float* T = targ + (size_t)plane * (Hh * Ww);

    // Zero the pad columns [134..137] of vsum once (weight-0 K lanes read them;
    // must be finite so 0 * x == 0).
    for (int idx = tid; idx < 5 * 16 * 4; idx += 128) {
        int s = idx >> 6, rem = idx & 63;
        vsum[s][rem >> 2][VCOLS + (rem & 3)] = 0.0f;
    }

    // ---- WMMA lane constants ----
    const int lane     = tid & 31;
    const int laneRow  = lane & 15;   // N (output col within tile) / M-row group
    const int laneHalf = lane >> 4;   // 0: K=0/1 ; 1: K=2/3  (f32 16x4 A layout)

    // Band matrix B(22x16), entries 1/49 on the 7-wide diagonal band, in 6 K-chunks.
    // B layout (4x16 f32): VGPR v, lane half h -> K = v + 2h, N = lane&15.
    v2f bmat[6];
#pragma unroll
    for (int kc = 0; kc < 6; ++kc) {
        int kg0 = kc * 4 + 0 + 2 * laneHalf;
        int kg1 = kc * 4 + 1 + 2 * laneHalf;
        int n   = laneRow;
        v2f b;
        b.x = (kg0 >= n && kg0 <= n + 6) ? (1.0f / 49.0f) : 0.0f;
        b.y = (kg1 >= n && kg1 <= n + 6) ? (1.0f / 49.0f) : 0.0f;
        bmat[kc] = b;
    }

    // ---- vertical sliding-window state (thread owns col c0, and c1 if tid<6) ----
    const int  c0   = tid;
    const int  c1   = tid + 128;
    const bool has2 = (tid < 6);
    float V0[5] = {0.f, 0.f, 0.f, 0.f, 0.f};
    float V1[5] = {0.f, 0.f, 0.f, 0.f, 0.f};

    auto loadPT = [&](int i, int c, float& p, float& t) {
        int gy = yb + i, gx = xb + c;
        if (gy < Hh && gx < Ww) { p = P[gy * Ww + gx]; t = T[gy * Ww + gx]; }
        else                    { p = 0.0f; t = 0.0f; }
    };
    auto enterRow = [&](int i) {
        float p, t;
        loadPT(i, c0, p, t);
        V0[0] += p; V0[1] += t; V0[2] += p * p; V0[3] += t * t; V0[4] += p * t;
        ring[i & 7][c0][0] = p; ring[i & 7][c0][1] = t;
        if (has2) {
            loadPT(i, c1, p, t);
            V1[0] += p; V1[1] += t; V1[2] += p * p; V1[3] += t * t; V1[4] += p * t;
            ring[i & 7][c1][0] = p; ring[i & 7][c1][1] = t;
        }
    };
    auto leaveRow = [&](int r) {
        float p = ring[r & 7][c0][0], t = ring[r & 7][c0][1];
        V0[0] -= p; V0[1] -= t; V0[2] -= p * p; V0[3] -= t * t; V0[4] -= p * t;
        if (has2) {
            p = ring[r & 7][c1][0]; t = ring[r & 7][c1][1];
            V1[0] -= p; V1[1] -= t; V1[2] -= p * p; V1[3] -= t * t; V1[4] -= p * t;
        }
    };

    // Prime first 6 input rows of the window.
    for (int i = 0; i < 6; ++i) enterRow(i);
    __syncthreads();   // pad zeros + (no hazard yet on vsum)

    float lsum = 0.0f;
    const float cn = 49.0f / 48.0f;        // sample-covariance norm N/(N-1)
    const float C1 = 4.0e-4f;              // (0.01*2)^2
    const float C2 = 3.6e-3f;              // (0.03*2)^2

    for (int b = 0; b < NBATCH; ++b) {
        // ---- vertical pass: produce 16 rows of 7-row sums for 5 stats ----
        for (int k = 0; k < 16; ++k) {
            int i = 6 + b * 16 + k;
            enterRow(i);                      // V now covers rows r..r+6, r=b*16+k
#pragma unroll
            for (int s = 0; s < 5; ++s) vsum[s][k][c0] = V0[s];
            if (has2) {
#pragma unroll
                for (int s = 0; s < 5; ++s) vsum[s][k][c1] = V1[s];
            }
            leaveRow(b * 16 + k);             // slide window down
        }
        __syncthreads();

        // ---- horizontal pass: banded matmul on the matrix unit ----
        const int wv = tid >> 5;              // wave id 0..3, each does 2 x-tiles
#pragma unroll
        for (int jt = 0; jt < 2; ++jt) {
            const int j = wv * 2 + jt;        // x-tile 0..7 (16 output cols each)
            v8f acc[5];
#pragma unroll
            for (int s = 0; s < 5; ++s) {
                v8f c = {0.f, 0.f, 0.f, 0.f, 0.f, 0.f, 0.f, 0.f};
#pragma unroll
                for (int kc = 0; kc < 6; ++kc) {
                    // A (16x4 f32): lane row holds M, VGPR pair holds K (+2 per half)
                    const float* ap = &vsum[s][laneRow][j * 16 + kc * 4 + laneHalf * 2];
                    v2f a = *(const v2f*)ap;
                    c = __builtin_amdgcn_wmma_f32_16x16x4_f32(
                            false, a, false, bmat[kc], (short)0, c, false, false);
                }
                acc[s] = c;
            }
            // SSIM formula per held element; C/D layout: M = v + 8*half, N = laneRow
#pragma unroll
            for (int v = 0; v < 8; ++v) {
                float ux = acc[0][v], uy = acc[1][v];
                float uxx = acc[2][v], uyy = acc[3][v], uxy = acc[4][v];
                float vx  = cn * (uxx - ux * ux);
                float vy  = cn * (uyy - uy * uy);
                float vxy = cn * (uxy - ux * uy);
                float num = (2.0f * ux * uy + C1) * (2.0f * vxy + C2);
                float den = (ux * ux + uy * uy + C1) * (vx + vy + C2);
                int oy = yb + b * 16 + v + 8 * laneHalf;
                int ox = xb + j * 16 + laneRow;
                if (oy < OUTW && ox < OUTW) lsum += num / den;
            }
        }
        __syncthreads();
    }

    // ---- block reduction + one atomic per block ----
    red[tid] = lsum;
    __syncthreads();
    for (int off = 64; off > 0; off >>= 1) {
        if (tid < off) red[tid] += red[tid + off];
        __syncthreads();
    }
    if (tid == 0) atomicAdd(accum, red[0]);
}

__global__ void ssim_finalize(const float* __restrict__ accum,
                              float* __restrict__ out) {
    // mean over 96 * 506 * 506 valid SSIM values; loss = 1 - mean
    out[0] = 1.0f - accum[0] * (1.0f / 24579456.0f);
}

extern "C" void kernel_launch(void* const* d_in, const int* in_sizes, int n_in,
                              void* d_out, int out_size, void* d_ws, size_t ws_size,
                              hipStream_t stream) {
    (void)in_sizes; (void)n_in; (void)out_size; (void)ws_size;
    const float* pred = (const float*)d_in[0];
    const float* targ = (const float*)d_in[1];
    float* out = (float*)d_out;
    float* acc = (float*)d_ws;

    hipMemsetAsync(acc, 0, sizeof(float), stream);
    dim3 grid(4, 8, 96);   // col-strips x row-blocks x (B*C) planes
    dim3 block(128);
    ssim_main<<<grid, block, 0, stream>>>(pred, targ, acc);
    ssim_finalize<<<1, 1, 0, stream>>>(acc, out);
}